// GAT_21869973471805
// MI455X (gfx1250) — compile-verified
//
#include <hip/hip_runtime.h>
#include <hip/hip_bf16.h>

// ---------------------------------------------------------------------------
// GAT (2-layer, PyG GATConv semantics) for MI455X / gfx1250.
// GEMMs use V_WMMA_F32_16X16X4_F32 (full fp32 matrix path).
// Edge softmax/scatter uses native f32/u32 global atomics (L2-resident set).
// ---------------------------------------------------------------------------

typedef float v2f __attribute__((ext_vector_type(2)));
typedef float v8f __attribute__((ext_vector_type(8)));

#define F_IN   128
#define HIDTOT 128   // H * HID
#define HEADS1 4
#define HID1   32
#define OUTC   32
#define NEG_SLOPE 0.2f

// ------------------------------ WMMA GEMM ----------------------------------
// C[nrows, KOUT] = A[nrows, KIN] @ W[KIN, KOUT]   (row-major, fp32)
// One 16x16 output tile per wave; blockDim = 32 * (KOUT/16) so all column
// tiles of a 16-row stripe share one block (A fragments reuse via cache).
template <int KIN, int KOUT>
__launch_bounds__(32 * (KOUT / 16))
__global__ void wmma_gemm_kernel(const float* __restrict__ A,
                                 const float* __restrict__ W,
                                 float* __restrict__ C,
                                 int nrows) {
    const int lane = threadIdx.x & 31;
    const int wave = threadIdx.x >> 5;          // column tile index
    const int row0 = blockIdx.x * 16;
    const int m    = lane & 15;                 // A row within tile
    const int kh   = lane >> 4;                 // K-half select (0/1)
    const int col  = wave * 16 + (lane & 15);   // B/C column

    const int arow = min(row0 + m, nrows - 1);
    const float* __restrict__ Arow = A + (long)arow * KIN;

    v8f acc = {};
#pragma unroll
    for (int k0 = 0; k0 < KIN; k0 += 4) {
        const int ka = k0 + 2 * kh;
        // A fragment: 16x4 f32, lane holds A[m, ka], A[m, ka+1]  (8B aligned)
        v2f a = *(const v2f*)(Arow + ka);
        // B fragment: 4x16 f32, lane holds W[ka, col], W[ka+1, col]
        v2f b;
        b.x = W[(long)ka * KOUT + col];
        b.y = W[(long)(ka + 1) * KOUT + col];
        acc = __builtin_amdgcn_wmma_f32_16x16x4_f32(
            /*neg_a=*/false, a, /*neg_b=*/false, b,
            /*c_mod=*/(short)0, acc, /*reuse_a=*/false, /*reuse_b=*/false);
    }
    // C/D layout: VGPR v holds row (v + 8*kh), column = lane&15 (per tile)
#pragma unroll
    for (int v = 0; v < 8; ++v) {
        const int r = row0 + v + 8 * kh;
        if (r < nrows) C[(long)r * KOUT + col] = acc[v];
    }
}

// --------------------------- helper kernels --------------------------------
__global__ void fill_zero_kernel(float* __restrict__ p, long n) {
    long i = (long)blockIdx.x * blockDim.x + threadIdx.x;
    if (i < n) p[i] = 0.0f;
}

__global__ void init_out_bias_kernel(float* __restrict__ out,
                                     const float* __restrict__ b,
                                     int n, int c) {
    long i = (long)blockIdx.x * blockDim.x + threadIdx.x;
    if (i < (long)n * c) out[i] = b[i % c];
}

// acc = relu(acc + bias)  (in-place; produces layer-2 input)
__global__ void bias_relu_kernel(float* __restrict__ acc,
                                 const float* __restrict__ b,
                                 long total, int c) {
    long i = (long)blockIdx.x * blockDim.x + threadIdx.x;
    if (i >= total) return;
    float v = acc[i] + b[i % c];
    acc[i] = v > 0.0f ? v : 0.0f;
}

// per-(node, head) attention dots: as = <h_nh, a_src_h>, ad = <h_nh, a_dst_h>
template <int H, int CH>
__global__ void alpha_kernel(const float* __restrict__ h,
                             const float* __restrict__ a_src,
                             const float* __restrict__ a_dst,
                             float* __restrict__ as, float* __restrict__ ad,
                             int n) {
    long idx = (long)blockIdx.x * blockDim.x + threadIdx.x;
    if (idx >= (long)n * H) return;
    const int node = (int)(idx / H);
    const int hd   = (int)(idx % H);
    const float* __restrict__ hp = h + (long)node * (H * CH) + hd * CH;
    float s = 0.0f, d = 0.0f;
#pragma unroll
    for (int c = 0; c < CH; ++c) {
        const float v = hp[c];
        s += v * a_src[hd * CH + c];
        d += v * a_dst[hd * CH + c];
    }
    as[idx] = s;
    ad[idx] = d;
}

// order-preserving float <-> uint key (for segment-max via atomic umax)
__device__ __forceinline__ unsigned fkey(float f) {
    unsigned b = __float_as_uint(f);
    return (b & 0x80000000u) ? ~b : (b | 0x80000000u);
}
__device__ __forceinline__ float funkey(unsigned k) {
    unsigned b = (k & 0x80000000u) ? (k ^ 0x80000000u) : ~k;
    return __uint_as_float(b);
}

__device__ __forceinline__ void edge_endpoints(const int* __restrict__ ei,
                                               long ep, int E, int& s, int& d) {
    if (ep < E) { s = ei[ep]; d = ei[(long)E + ep]; }
    else        { s = d = (int)(ep - E); }
}

// pass 1: e = leaky_relu(as[src]+ad[dst]); stash e; segment max over dst
template <int H>
__global__ void edge_logits_max_kernel(const int* __restrict__ ei, int E, int n,
                                       const float* __restrict__ as,
                                       const float* __restrict__ ad,
                                       float* __restrict__ ebuf,
                                       unsigned* __restrict__ emax) {
    long ep = (long)blockIdx.x * blockDim.x + threadIdx.x;
    if (ep >= (long)E + n) return;
    int s, d;
    edge_endpoints(ei, ep, E, s, d);
#pragma unroll
    for (int hd = 0; hd < H; ++hd) {
        float e = as[(long)s * H + hd] + ad[(long)d * H + hd];
        e = e > 0.0f ? e : NEG_SLOPE * e;
        ebuf[ep * H + hd] = e;
        atomicMax(&emax[(long)d * H + hd], fkey(e));
    }
}

// pass 2: ex = exp(e - emax[dst]); stash ex; segment sum of ex over dst
template <int H>
__global__ void edge_expsum_kernel(const int* __restrict__ ei, int E, int n,
                                   const unsigned* __restrict__ emax,
                                   float* __restrict__ ebuf,
                                   float* __restrict__ denom) {
    long ep = (long)blockIdx.x * blockDim.x + threadIdx.x;
    if (ep >= (long)E + n) return;
    int s, d;
    edge_endpoints(ei, ep, E, s, d);
#pragma unroll
    for (int hd = 0; hd < H; ++hd) {
        float ex = __expf(ebuf[ep * H + hd] - funkey(emax[(long)d * H + hd]));
        ebuf[ep * H + hd] = ex;
        atomicAdd(&denom[(long)d * H + hd], ex);
    }
}

// pass 3: out[dst, ch] += h[src, ch] * (ex / denom[dst, head(ch)])
template <int CT, int H>
__global__ void edge_scatter_kernel(const int* __restrict__ ei, int E, int n,
                                    const float* __restrict__ h,
                                    const float* __restrict__ ebuf,
                                    const float* __restrict__ denom,
                                    float* __restrict__ out) {
    long t = (long)blockIdx.x * blockDim.x + threadIdx.x;
    const long EP = (long)E + n;
    if (t >= EP * CT) return;
    const long ep = t / CT;           // CT is a power of two -> shifts
    const int  ch = (int)(t % CT);
    int s, d;
    edge_endpoints(ei, ep, E, s, d);
    const int hd = ch / (CT / H);
    const float alpha = ebuf[ep * H + hd] / denom[(long)d * H + hd];
    atomicAdd(&out[(long)d * CT + ch], h[(long)s * CT + ch] * alpha);
}

// ------------------------------- launch ------------------------------------
extern "C" void kernel_launch(void* const* d_in, const int* in_sizes, int n_in,
                              void* d_out, int out_size, void* d_ws, size_t ws_size,
                              hipStream_t stream) {
    const float* x      = (const float*)d_in[0];
    const int*   ei     = (const int*)d_in[1];
    const float* W1     = (const float*)d_in[2];
    const float* a_src1 = (const float*)d_in[3];
    const float* a_dst1 = (const float*)d_in[4];
    const float* b1     = (const float*)d_in[5];
    const float* W2     = (const float*)d_in[6];
    const float* a_src2 = (const float*)d_in[7];
    const float* a_dst2 = (const float*)d_in[8];
    const float* b2     = (const float*)d_in[9];
    float*       out    = (float*)d_out;

    const int N  = in_sizes[0] / F_IN;   // 100000
    const int E  = in_sizes[1] / 2;      // 1600000
    const long EP = (long)E + N;         // with self loops

    // workspace layout (floats)
    float* w = (float*)d_ws;
    float*    h1     = w;                    w += (long)N * HIDTOT;  // gemm1 out / reused as h2
    float*    acc1   = w;                    w += (long)N * HIDTOT;  // layer-1 aggregate -> relu input
    float*    as1    = w;                    w += (long)N * HEADS1;
    float*    ad1    = w;                    w += (long)N * HEADS1;
    unsigned* emax1  = (unsigned*)w;         w += (long)N * HEADS1;
    float*    den1   = w;                    w += (long)N * HEADS1;
    float*    ebuf1  = w;                    w += EP * HEADS1;
    float*    as2    = w;                    w += N;
    float*    ad2    = w;                    w += N;
    unsigned* emax2  = (unsigned*)w;         w += N;
    float*    den2   = w;                    w += N;
    float*    ebuf2  = w;                    w += EP;

    const int  B      = 256;
    const long nh1    = (long)N * HEADS1;
    const long nacc   = (long)N * HIDTOT;
    const int  rowtiles = (N + 15) / 16;
    auto blks = [](long total, int b) { return (unsigned)((total + b - 1) / b); };

    // ---------------- layer 1 ----------------
    fill_zero_kernel<<<blks(nacc, B), B, 0, stream>>>(acc1, nacc);
    fill_zero_kernel<<<blks(nh1, B), B, 0, stream>>>((float*)emax1, nh1); // key 0 == -inf sentinel
    fill_zero_kernel<<<blks(nh1, B), B, 0, stream>>>(den1, nh1);

    wmma_gemm_kernel<F_IN, HIDTOT>
        <<<rowtiles, 32 * (HIDTOT / 16), 0, stream>>>(x, W1, h1, N);

    alpha_kernel<HEADS1, HID1>
        <<<blks(nh1, B), B, 0, stream>>>(h1, a_src1, a_dst1, as1, ad1, N);

    edge_logits_max_kernel<HEADS1>
        <<<blks(EP, B), B, 0, stream>>>(ei, E, N, as1, ad1, ebuf1, emax1);
    edge_expsum_kernel<HEADS1>
        <<<blks(EP, B), B, 0, stream>>>(ei, E, N, emax1, ebuf1, den1);
    edge_scatter_kernel<HIDTOT, HEADS1>
        <<<blks(EP * HIDTOT, B), B, 0, stream>>>(ei, E, N, h1, ebuf1, den1, acc1);

    bias_relu_kernel<<<blks(nacc, B), B, 0, stream>>>(acc1, b1, nacc, HIDTOT);

    // ---------------- layer 2 ----------------
    fill_zero_kernel<<<blks(N, B), B, 0, stream>>>((float*)emax2, N);
    fill_zero_kernel<<<blks(N, B), B, 0, stream>>>(den2, N);

    wmma_gemm_kernel<HIDTOT, OUTC>
        <<<rowtiles, 32 * (OUTC / 16), 0, stream>>>(acc1, W2, h1 /*reused as h2*/, N);

    alpha_kernel<1, OUTC>
        <<<blks(N, B), B, 0, stream>>>(h1, a_src2, a_dst2, as2, ad2, N);

    edge_logits_max_kernel<1>
        <<<blks(EP, B), B, 0, stream>>>(ei, E, N, as2, ad2, ebuf2, emax2);
    edge_expsum_kernel<1>
        <<<blks(EP, B), B, 0, stream>>>(ei, E, N, emax2, ebuf2, den2);

    init_out_bias_kernel<<<blks((long)N * OUTC, B), B, 0, stream>>>(out, b2, N, OUTC);
    edge_scatter_kernel<OUTC, 1>
        <<<blks(EP * OUTC, B), B, 0, stream>>>(ei, E, N, h1, ebuf2, den2, out);
}